// MultiFreqVMEncoder_24111946399762
// MI455X (gfx1250) — compile-verified
//
#include <hip/hip_runtime.h>
#include <stdint.h>

// ---------------------------------------------------------------------------
// MultiFreqVMEncoder for MI455X (gfx1250)
//   S=6 scales, XD=3, C=8 channels, R=64, B = S*2*XD = 36, out = (n, 288)
//
// Strategy (bandwidth-bound, no matmul structure -> no WMMA; use TDM + LDS):
//   kernel 1: lineTab[row=b*8+c][y] = 0.5*(feat[row][y][31] + feat[row][y][32])
//             (gx==0 => x-lerp of the 1D sample is a constant 0.5/0.5 blend)
//   kernel 2: per 256-thread block, TDM-load lineTab into LDS with HW pad
//             (row stride 64 -> 68 dwords, bank-conflict free), per-wave
//             32-point tiles: phase A = per-lane sincos latents -> LDS,
//             phase B = output-major loop giving perfectly coalesced stores.
// ---------------------------------------------------------------------------

#define NS     6
#define NXD    3
#define NC     8
#define RR     64
#define NB     36            // NS * 2 * NXD
#define ROWS   288           // NB * NC
#define RSTR   68            // 64 data dwords + 4 pad dwords (TDM pad)
#define OUTD   288           // NC * NB
#define LINE_ELEMS (ROWS * RR)   // 18432 floats in workspace

#if __has_builtin(__builtin_amdgcn_tensor_load_to_lds) && \
    __has_builtin(__builtin_amdgcn_s_wait_tensorcnt)
#define HAVE_TDM 1
#else
#define HAVE_TDM 0
#endif

typedef unsigned int v4u __attribute__((ext_vector_type(4)));
typedef int          v8i __attribute__((ext_vector_type(8)));
typedef int          v4i __attribute__((ext_vector_type(4)));

__device__ __forceinline__ int iclamp(int v, int lo, int hi) {
    return v < lo ? lo : (v > hi ? hi : v);
}

// ---------------------------------------------------------------------------
// Kernel 1: collapse the 1-D line features (gx = 0 -> wx = 0.5 exactly).
// ws[row*64 + y] = 0.5*(features[(row*64+y)*64 + 31] + features[...+32])
// ---------------------------------------------------------------------------
__global__ void line_precompute_kernel(const float* __restrict__ feat,
                                       float* __restrict__ ws) {
    int t = blockIdx.x * blockDim.x + threadIdx.x;
    if (t < LINE_ELEMS) {
        size_t o = (size_t)t * RR;
        ws[t] = 0.5f * (feat[o + 31] + feat[o + 32]);
    }
}

// ---------------------------------------------------------------------------
// Kernel 2: main encoder.
// ---------------------------------------------------------------------------
__global__ void __launch_bounds__(256, 2)
encoder_kernel(const float* __restrict__ x,
               const float* __restrict__ feat,      // only used if lineG==null
               const float* __restrict__ feat2d,
               const float* __restrict__ lineG,     // precomputed line table
               float* __restrict__ out,
               int n) {
    __shared__ float lineLds[ROWS * RSTR];   // 78,336 B, padded rows
    __shared__ float latLds[256 * NB];       // 36,864 B, per-lane latents

    const int tid = threadIdx.x;
    const int wv  = tid >> 5;
    const int ln  = tid & 31;

    // ---------------- stage the line table into LDS ----------------
    if (lineG != nullptr) {
#if HAVE_TDM
        if (tid < 32) {
            // Tensor DMA descriptor: 2D tensor 64 x 288 dwords, tile = whole
            // tensor, pad 4 dwords after every 64 dwords -> LDS row stride 68.
            uint64_t ga = (uint64_t)(uintptr_t)lineG;
            uint32_t la = (uint32_t)(uintptr_t)(&lineLds[0]);
            v4u g0;
            g0[0] = 1u;                                   // count=1 (valid)
            g0[1] = la;                                   // lds_addr
            g0[2] = (uint32_t)ga;                         // global_addr[31:0]
            g0[3] = (uint32_t)((ga >> 32) & 0x1FFFFFFu)   // global_addr[56:32]
                  | (2u << 30);                           // type = 2 (image)
            v8i g1;
            g1[0] = (int)((2u << 16)        // data_size = 4B
                        | (1u << 20)        // pad_enable
                        | (5u << 22)        // pad_interval: 64 dwords
                        | (3u << 25));      // pad_amount:   4 dwords
            g1[1] = (int)(64u << 16);       // tensor_dim0[15:0] = 64
            g1[2] = (int)(288u << 16);      // tensor_dim1[15:0] = 288
            g1[3] = (int)(64u << 16);       // tile_dim0 = 64
            g1[4] = (int)288u;              // tile_dim1 = 288, tile_dim2 = 0
            g1[5] = (int)64u;               // tensor_dim0_stride = 64
            g1[6] = (int)(18432u << 16);    // tensor_dim1_stride[15:0]
            g1[7] = 0;
            v4i gz4 = {0, 0, 0, 0};
            v8i gz8 = {0, 0, 0, 0, 0, 0, 0, 0};
            // 6-arg form (amdgpu-toolchain / clang-23 + therock-10.0 headers)
            __builtin_amdgcn_tensor_load_to_lds(g0, g1, gz4, gz4, gz8, 0);
            __builtin_amdgcn_s_wait_tensorcnt(0);
        }
#else
        // Async global->LDS copy, 16B chunks, padded destination rows.
        for (int i = 0; i < (ROWS * 16 + 255) / 256; ++i) {
            int chunk = tid + i * 256;
            if (chunk < ROWS * 16) {
                int row = chunk >> 4, piece = chunk & 15;
                uint64_t src = (uint64_t)(uintptr_t)(lineG + (row * RR + piece * 4));
                uint32_t dst = (uint32_t)(uintptr_t)(&lineLds[row * RSTR + piece * 4]);
                asm volatile("global_load_async_to_lds_b128 %0, %1, off"
                             :: "v"(dst), "v"(src) : "memory");
            }
        }
        asm volatile("s_wait_asynccnt 0" ::: "memory");
#endif
    } else {
        // Workspace too small: build line table in-kernel from raw features.
        for (int e = tid; e < LINE_ELEMS; e += 256) {
            int row = e >> 6, y = e & 63;
            size_t o = (size_t)e * RR;
            lineLds[row * RSTR + y] = 0.5f * (feat[o + 31] + feat[o + 32]);
        }
    }

    // ---------------- phase A: per-lane latents ----------------
    const int p = blockIdx.x * 256 + tid;        // this lane's point
    float xv0 = 0.f, xv1 = 0.f, xv2 = 0.f;
    if (p < n) {
        xv0 = x[(size_t)p * 3 + 0];
        xv1 = x[(size_t)p * 3 + 1];
        xv2 = x[(size_t)p * 3 + 2];
    }
    {
        float* lat = latLds + tid * NB;
        float sc = 1.f;
        #pragma unroll
        for (int s = 0; s < NS; ++s) {
            float sv, cv;
            __sincosf(xv0 * sc, &sv, &cv);
            lat[s * 6 + 0] = sv; lat[s * 6 + 3] = cv;
            __sincosf(xv1 * sc, &sv, &cv);
            lat[s * 6 + 1] = sv; lat[s * 6 + 4] = cv;
            __sincosf(xv2 * sc, &sv, &cv);
            lat[s * 6 + 2] = sv; lat[s * 6 + 5] = cv;
            sc *= 2.f;
        }
    }
    __syncthreads();

    // ---------------- phase B: output-major, coalesced stores ----------------
    // Wave `wv` owns points [base, base+32). Lane l produces output element
    // idx = g*32 + l for each point -> every store is one contiguous 128B line.
    const int   base = blockIdx.x * 256 + wv * 32;
    const float* latp = latLds + (wv * 32) * NB;

    #pragma unroll
    for (int g = 0; g < 9; ++g) {
        const int idx = g * 32 + ln;          // 0..287
        const int c   = idx / NB;             // channel
        const int r   = idx - c * NB;         // = b = s*6 + j*3 + d
        const int d   = r % 3;
        const int rb  = r - d;
        const int rp0 = rb + ((d + 1) % 3);   // pairs[d][0]
        const int rp1 = rb + ((d + 2) % 3);   // pairs[d][1]
        const float* __restrict__ line = lineLds + (r * NC + c) * RSTR;
        const float* __restrict__ img  = feat2d + (size_t)(r * NC + c) * (RR * RR);

        for (int pp = 0; pp < 32; ++pp) {
            const float* lrow = latp + pp * NB;
            const float lr  = lrow[r];        // gy for line + additive term
            const float gx2 = lrow[rp0];
            const float gy2 = lrow[rp1];

            // 1-D line sample (x already collapsed at wx=0.5)
            float fy  = (lr + 1.f) * 31.5f;
            float f0  = floorf(fy);
            float wy  = fy - f0;
            int   iy0 = iclamp((int)f0, 0, 63);
            int   iy1 = iy0 + 1 > 63 ? 63 : iy0 + 1;
            float l0  = line[iy0];
            float fsv = l0 + wy * (line[iy1] - l0);

            // 2-D bilinear from plane features (L2-resident gathers)
            float fx  = (gx2 + 1.f) * 31.5f;
            float fyy = (gy2 + 1.f) * 31.5f;
            float fx0 = floorf(fx);
            float fy0b = floorf(fyy);
            float wx  = fx - fx0;
            float wy2 = fyy - fy0b;
            int ix0 = iclamp((int)fx0, 0, 63);
            int ix1 = ix0 + 1 > 63 ? 63 : ix0 + 1;
            int jy0 = iclamp((int)fy0b, 0, 63);
            int jy1 = jy0 + 1 > 63 ? 63 : jy0 + 1;
            const float* r0 = img + jy0 * RR;
            const float* r1 = img + jy1 * RR;
            float v00 = r0[ix0], v01 = r0[ix1];
            float v10 = r1[ix0], v11 = r1[ix1];
            float ta  = v00 + wx * (v01 - v00);
            float tb  = v10 + wx * (v11 - v10);
            float f2v = ta + wy2 * (tb - ta);

            int pg = base + pp;
            if (pg < n) out[(size_t)pg * OUTD + idx] = fsv * f2v + lr;
        }
    }
}

// ---------------------------------------------------------------------------
extern "C" void kernel_launch(void* const* d_in, const int* in_sizes, int n_in,
                              void* d_out, int out_size, void* d_ws, size_t ws_size,
                              hipStream_t stream) {
    const float* x      = (const float*)d_in[0];
    const float* feat   = (const float*)d_in[1];
    const float* feat2d = (const float*)d_in[2];
    float*       out    = (float*)d_out;
    const int n = in_sizes[0] / 3;

    float* lineG = nullptr;
    if (ws_size >= LINE_ELEMS * sizeof(float)) lineG = (float*)d_ws;

    if (lineG) {
        line_precompute_kernel<<<(LINE_ELEMS + 255) / 256, 256, 0, stream>>>(feat, lineG);
    }
    const int blocks = (n + 255) / 256;     // 256 points per block (8 waves x 32)
    encoder_kernel<<<blocks, 256, 0, stream>>>(x, feat, feat2d, lineG, out, n);
}